// GraphSAGE_20976620274055
// MI455X (gfx1250) — compile-verified
//
#include <hip/hip_runtime.h>
#include <hip/hip_bf16.h>

#define NNODES 40000
#define NEDGES 640000
#define DIM    128

typedef __attribute__((ext_vector_type(2))) float v2f;
typedef __attribute__((ext_vector_type(8))) float v8f;

// ---------------------------------------------------------------------------
// Degree: one f32 atomicAdd per edge (computed once, shared by both layers)
// ---------------------------------------------------------------------------
__global__ void sage_degree(const int* __restrict__ dst, float* __restrict__ deg) {
    int e = blockIdx.x * blockDim.x + threadIdx.x;
    if (e < NEDGES) atomicAdd(&deg[dst[e]], 1.0f);
}

// ---------------------------------------------------------------------------
// Scatter-add: one wave32 per edge. Lane l moves columns [4l, 4l+4) as a
// float4 (coalesced 512B read per wave) and atomically accumulates into
// sums[dst]. Sums buffer (20.5 MB) is L2-resident -> atomics resolve in L2.
// ---------------------------------------------------------------------------
__global__ void sage_scatter(const float* __restrict__ H,
                             const int*   __restrict__ src,
                             const int*   __restrict__ dst,
                             float*       __restrict__ sums) {
    int gid  = blockIdx.x * blockDim.x + threadIdx.x;
    int e    = gid >> 5;
    int lane = gid & 31;
    if (e >= NEDGES) return;
    const float* s = H    + (size_t)src[e] * DIM + lane * 4;
    float*       d = sums + (size_t)dst[e] * DIM + lane * 4;
    float4 v = *(const float4*)s;
    atomicAdd(d + 0, v.x);
    atomicAdd(d + 1, v.y);
    atomicAdd(d + 2, v.z);
    atomicAdd(d + 3, v.w);
}

// ---------------------------------------------------------------------------
// Fused dual-GEMM tile kernel using V_WMMA_F32_16X16X4_F32 (full f32
// precision; GEMM is ~5 GFLOP total, far below any roofline, so keep the
// reference's f32 semantics).
//
//   out[m,n] = sum_k H[m,k]*Ws[k,n] + (sums[m,k]*inv_deg[m])*Wn[k,n] + b[n]
//
// Block = 256 threads = 8 waves; wave w owns N-tile w (N=128 -> 8 tiles),
// blockIdx.x owns M-tile (40000/16 = 2500 tiles). K-loop: 32 steps of 4.
//
// Fragment layouts (ISA 7.12.2, wave32):
//   A 16x4 f32 : lane = 16*kh + m ; v0 = A[m][2kh], v1 = A[m][2kh+1]
//   B  4x16 f32: lane = 16*kh + n ; v0 = B[2kh][n], v1 = B[2kh+1][n]
//   C/D 16x16  : lane = 16*kh + n ; vi = C[i + 8*kh][n]
// No divergence anywhere in this kernel -> EXEC all-1s (WMMA requirement).
// ---------------------------------------------------------------------------
__global__ void sage_gemm(const float* __restrict__ H,
                          const float* __restrict__ Msum,
                          const float* __restrict__ deg,
                          const float* __restrict__ Wself,
                          const float* __restrict__ Wneigh,
                          const float* __restrict__ bias,
                          float*       __restrict__ out,
                          int do_relu) {
    const int wave = threadIdx.x >> 5;   // 0..7 -> N tile
    const int lane = threadIdx.x & 31;
    const int kh   = lane >> 4;          // half-select
    const int l16  = lane & 15;
    const int m0   = blockIdx.x * 16;
    const int n0   = wave * 16;

    const int   rowA = m0 + l16;                 // A-fragment row for this lane
    const float inv  = 1.0f / fmaxf(deg[rowA], 1.0f);

    const float* hrow = H    + (size_t)rowA * DIM;
    const float* srow = Msum + (size_t)rowA * DIM;

    v8f acc = {};
    #pragma unroll 4
    for (int k = 0; k < DIM; k += 4) {
        const int ka = k + 2 * kh;
        v2f aS = { hrow[ka],       hrow[ka + 1] };
        v2f aN = { srow[ka] * inv, srow[ka + 1] * inv };
        v2f bS = { Wself [(size_t)ka * DIM + n0 + l16],
                   Wself [(size_t)(ka + 1) * DIM + n0 + l16] };
        v2f bN = { Wneigh[(size_t)ka * DIM + n0 + l16],
                   Wneigh[(size_t)(ka + 1) * DIM + n0 + l16] };
        acc = __builtin_amdgcn_wmma_f32_16x16x4_f32(
                  false, aS, false, bS, (short)0, acc, false, false);
        acc = __builtin_amdgcn_wmma_f32_16x16x4_f32(
                  false, aN, false, bN, (short)0, acc, false, false);
    }

    const float bn = bias[n0 + l16];
    #pragma unroll
    for (int i = 0; i < 8; ++i) {
        float v = acc[i] + bn;
        if (do_relu) v = fmaxf(v, 0.0f);
        out[(size_t)(m0 + i + 8 * kh) * DIM + (n0 + l16)] = v;
    }
}

// ---------------------------------------------------------------------------
extern "C" void kernel_launch(void* const* d_in, const int* in_sizes, int n_in,
                              void* d_out, int out_size, void* d_ws, size_t ws_size,
                              hipStream_t stream) {
    const float* x   = (const float*)d_in[0];
    const int*   src = (const int*)  d_in[1];
    const int*   dst = (const int*)  d_in[2];
    const float* Ws1 = (const float*)d_in[3];
    const float* Wn1 = (const float*)d_in[4];
    const float* b1  = (const float*)d_in[5];
    const float* Ws2 = (const float*)d_in[6];
    const float* Wn2 = (const float*)d_in[7];
    const float* b2  = (const float*)d_in[8];
    float*       out = (float*)d_out;

    char*  ws   = (char*)d_ws;
    float* sums = (float*)ws;                                        // 20.48 MB
    float* deg  = (float*)(ws + (size_t)NNODES * DIM * sizeof(float));
    float* h1   = (float*)(ws + (size_t)NNODES * DIM * sizeof(float)
                              + (size_t)NNODES * sizeof(float));     // 20.48 MB

    const int scatter_blocks = (NEDGES * 32) / 256;   // 80000, exact
    const int gemm_blocks    = NNODES / 16;           // 2500, exact

    // ---- Layer 1 ----
    hipMemsetAsync(sums, 0, (size_t)NNODES * DIM * sizeof(float), stream);
    hipMemsetAsync(deg,  0, (size_t)NNODES * sizeof(float),       stream);
    sage_degree <<<(NEDGES + 255) / 256, 256, 0, stream>>>(dst, deg);
    sage_scatter<<<scatter_blocks, 256, 0, stream>>>(x, src, dst, sums);
    sage_gemm   <<<gemm_blocks, 256, 0, stream>>>(x, sums, deg, Ws1, Wn1, b1, h1, 1);

    // ---- Layer 2 ----
    hipMemsetAsync(sums, 0, (size_t)NNODES * DIM * sizeof(float), stream);
    sage_scatter<<<scatter_blocks, 256, 0, stream>>>(h1, src, dst, sums);
    sage_gemm   <<<gemm_blocks, 256, 0, stream>>>(h1, sums, deg, Ws2, Wn2, b2, out, 0);
}